// FTT_Layer_73091753443635
// MI455X (gfx1250) — compile-verified
//
#include <hip/hip_runtime.h>
#include <hip/hip_bf16.h>

// ---------------------------------------------------------------------------
// FTT layer on MI455X (gfx1250, wave32).
//   Phase 1: fused streaming reduction  V0[3], V1..V3[3x3]  (HBM-bound, ~124MB)
//            256 blocks x 256 thr, 4 grid-stride iters/thread so the 30-way
//            shuffle-reduction epilogue is amortized over 4x the bytes.
//   Phase 2: f = V0 @ V1 @ V2 @ V3  via V_WMMA_F32_16X16X4_F32 (one wave)
//   Phase 3: out[s] = sum_r f[r] * TT4[r,s,0]               (HBM-bound, ~16MB)
// Whole working set (~140MB) fits in the 192MB L2 -> keep default RT cache
// hints so graph-replayed launches hit L2 instead of HBM.
// ---------------------------------------------------------------------------

typedef float  f32x2 __attribute__((ext_vector_type(2)));
typedef float  f32x4 __attribute__((ext_vector_type(4)));
typedef float  f32x8 __attribute__((ext_vector_type(8)));

// Accumulate acc[q] += sum over 4 consecutive s of z[s]*M[s,q], q=0..2,
// where M rows are 3 contiguous floats per s (12 contiguous floats total,
// 48B = three aligned b128 loads).
__device__ __forceinline__ void acc12(const float* __restrict__ base,
                                      f32x4 zz, float* __restrict__ acc) {
    const f32x4* p = (const f32x4*)base;
    f32x4 u0 = p[0], u1 = p[1], u2 = p[2];
    float v[12];
    v[0]=u0[0]; v[1]=u0[1]; v[2] =u0[2]; v[3] =u0[3];
    v[4]=u1[0]; v[5]=u1[1]; v[6] =u1[2]; v[7] =u1[3];
    v[8]=u2[0]; v[9]=u2[1]; v[10]=u2[2]; v[11]=u2[3];
#pragma unroll
    for (int i = 0; i < 4; ++i)
#pragma unroll
        for (int q = 0; q < 3; ++q)
            acc[q] += zz[i] * v[i * 3 + q];
}

// ---------------------------------------------------------------------------
// Phase 1: 30 parallel dot products over S. ws[0..2]=V0, ws[3+9k ..] = V_{k+1}
// ---------------------------------------------------------------------------
__global__ void ftt_reduce_kernel(const float* __restrict__ z,
                                  const float* __restrict__ T0,
                                  const float* __restrict__ T1,
                                  const float* __restrict__ T2,
                                  const float* __restrict__ T3,
                                  float* __restrict__ ws, int S) {
    float a[30];
#pragma unroll
    for (int i = 0; i < 30; ++i) a[i] = 0.f;

    const size_t stride = (size_t)gridDim.x * blockDim.x * 4;
    const size_t tid    = (size_t)blockIdx.x * blockDim.x + threadIdx.x;
    const float* cores[3] = {T1, T2, T3};

    for (size_t s4 = tid * 4; s4 < (size_t)S; s4 += stride) {
        f32x4 zz = *(const f32x4*)(z + s4);
        acc12(T0 + s4 * 3, zz, a);           // V0: TT0[0,s,q] at s*3+q
#pragma unroll
        for (int k = 0; k < 3; ++k)
#pragma unroll
            for (int r = 0; r < 3; ++r)       // TTk[r,s,q] at (r*S+s)*3+q
                acc12(cores[k] + ((size_t)r * S + s4) * 3, zz,
                      a + 3 + k * 9 + r * 3);
    }

    // wave32 tree reduction of all 30 accumulators (runs once per thread;
    // amortized over 4 streaming iterations)
#pragma unroll
    for (int i = 0; i < 30; ++i) {
        float v = a[i];
#pragma unroll
        for (int off = 16; off >= 1; off >>= 1)
            v += __shfl_xor(v, off, 32);
        a[i] = v;
    }

    __shared__ float sacc[30];
    if (threadIdx.x < 30) sacc[threadIdx.x] = 0.f;
    __syncthreads();
    if ((threadIdx.x & 31) == 0) {
#pragma unroll
        for (int i = 0; i < 30; ++i) atomicAdd(&sacc[i], a[i]);
    }
    __syncthreads();
    if (threadIdx.x < 30) atomicAdd(&ws[threadIdx.x], sacc[threadIdx.x]);
}

// ---------------------------------------------------------------------------
// Phase 2: f = V0 @ V1 @ V2 @ V3 with three V_WMMA_F32_16X16X4_F32.
// One wave (32 threads), EXEC all ones at each WMMA.
//   A 16x4 f32: lane m (m<16) holds {A[m][0], A[m][1]}; lane m+16 holds
//               {A[m][2], A[m][3]}.
//   B 4x16 f32: VGPR0 = rows 0 (lanes 0-15) / 2 (lanes 16-31),
//               VGPR1 = rows 1 / 3.
//   D 16x16 f32: VGPR j = row j (lanes 0-15) / row j+8 (lanes 16-31).
// Result row 0 -> c[0] of lanes 0..2.
// ---------------------------------------------------------------------------
__global__ void ftt_chain_wmma_kernel(float* __restrict__ ws) {
    const int  lane = threadIdx.x;
    const bool hi   = lane >= 16;
    const int  l    = lane & 15;
    __shared__ float fbuf[3];

    if (lane < 3) fbuf[lane] = ws[lane];   // f := V0 (1x3)
    __syncthreads();

#pragma unroll
    for (int k = 0; k < 3; ++k) {
        const float* M = ws + 3 + k * 9;   // V_{k+1}, row-major 3x3

        f32x2 av; av[0] = 0.f; av[1] = 0.f;     // A: row 0 = f, rest 0
        if (lane == 0)  { av[0] = fbuf[0]; av[1] = fbuf[1]; }
        if (lane == 16) { av[0] = fbuf[2]; }

        f32x2 bv; bv[0] = 0.f; bv[1] = 0.f;     // B: rows 0..2 = M, row 3 = 0
        if (l < 3) {
            if (!hi) { bv[0] = M[0 * 3 + l]; bv[1] = M[1 * 3 + l]; }
            else     { bv[0] = M[2 * 3 + l]; }
        }

        f32x8 c = {};
        c = __builtin_amdgcn_wmma_f32_16x16x4_f32(
                /*neg_a=*/false, av, /*neg_b=*/false, bv,
                /*c_mod=*/(short)0, c, /*reuse_a=*/false, /*reuse_b=*/false);

        __syncthreads();                    // all lanes done reading fbuf
        if (!hi && l < 3) fbuf[l] = c[0];   // D[0][n], n = lane 0..2
        __syncthreads();
    }

    if (lane < 3) ws[30 + lane] = fbuf[lane];
}

// ---------------------------------------------------------------------------
// Phase 3: out[s] = f0*T4[0*S+s] + f1*T4[1*S+s] + f2*T4[2*S+s]
// ---------------------------------------------------------------------------
__global__ void ftt_expand_kernel(const float* __restrict__ T4,
                                  const float* __restrict__ f,
                                  float* __restrict__ out, int S) {
    const size_t s4 = ((size_t)blockIdx.x * blockDim.x + threadIdx.x) * 4;
    if (s4 >= (size_t)S) return;
    const float f0 = f[0], f1 = f[1], f2 = f[2];
    f32x4 r0 = *(const f32x4*)(T4 + s4);
    f32x4 r1 = *(const f32x4*)(T4 + (size_t)S + s4);
    f32x4 r2 = *(const f32x4*)(T4 + 2 * (size_t)S + s4);
    f32x4 r  = f0 * r0 + f1 * r1 + f2 * r2;
    *(f32x4*)(out + s4) = r;
}

extern "C" void kernel_launch(void* const* d_in, const int* in_sizes, int n_in,
                              void* d_out, int out_size, void* d_ws, size_t ws_size,
                              hipStream_t stream) {
    const float* z  = (const float*)d_in[0];
    const float* T0 = (const float*)d_in[1];
    const float* T1 = (const float*)d_in[2];
    const float* T2 = (const float*)d_in[3];
    const float* T3 = (const float*)d_in[4];
    const float* T4 = (const float*)d_in[5];
    float* out = (float*)d_out;
    float* ws  = (float*)d_ws;
    const int S = in_sizes[0];   // 1024*1024

    // Zero the 30 accumulators + 3 result slots every call (graph-replay safe).
    hipMemsetAsync(ws, 0, 33 * sizeof(float), stream);

    const int threads = 256;

    // Phase 1: 256 blocks -> 65536 threads -> 4 grid-stride iterations each
    // (2048 waves in flight; each iter issues 31 b128 loads, plenty of MLP
    // to saturate 23.3 TB/s while amortizing the reduction epilogue).
    ftt_reduce_kernel<<<256, threads, 0, stream>>>(z, T0, T1, T2, T3, ws, S);

    ftt_chain_wmma_kernel<<<1, 32, 0, stream>>>(ws);

    // Phase 3: pure streaming, 1 float4 per thread.
    const int blocks3 = (S / 4 + threads - 1) / threads;   // 1024 blocks
    ftt_expand_kernel<<<blocks3, threads, 0, stream>>>(T4, ws + 30, out, S);
}